// LIFSpike_1159641169969
// MI455X (gfx1250) — compile-verified
//
#include <hip/hip_runtime.h>

// LIF temporal scan, x: [B=32, T=8, C=128, H=32, W=32] f32 -> spikes same shape.
// Memory-bound streaming kernel: 256 MB total traffic, floor ~11.5us @ 23.3 TB/s.
// One thread owns one float4 of a spatial slice and walks the T axis
// (stride C*H*W = 131072 elems = 512 KB) with 8 in-flight global_load_b128 (NT),
// runs the serial 8-step recurrence in registers, and streams 8 global_store_b128 (NT).

typedef float v4f __attribute__((ext_vector_type(4)));

constexpr int   T_STEPS  = 8;
constexpr int   SVEC     = (128 * 32 * 32) / 4;   // float4 per (b,t) slice = 32768 = 2^15
constexpr int   SVEC_LOG = 15;
constexpr float TAU      = 0.25f;
constexpr float THRESH   = 0.5f;

__global__ __launch_bounds__(256) void lif_scan_kernel(const v4f* __restrict__ x,
                                                       v4f* __restrict__ out,
                                                       int nthreads) {
    const int tid = blockIdx.x * blockDim.x + threadIdx.x;
    if (tid >= nthreads) return;

    const int b = tid >> SVEC_LOG;          // batch index
    const int s = tid & (SVEC - 1);         // float4 index within the spatial slice
    const size_t base = (size_t)b * (size_t)(T_STEPS * SVEC) + (size_t)s;

    // Issue all 8 time-step loads up front: 8 outstanding b128 NT loads per lane
    // gives the memory system deep MLP to hide HBM latency.
    v4f xv[T_STEPS];
#pragma unroll
    for (int t = 0; t < T_STEPS; ++t) {
        xv[t] = __builtin_nontemporal_load(&x[base + (size_t)t * SVEC]);
    }

    // Serial LIF recurrence over T, vectorized over the 4 packed spatial elems.
    v4f m;
    m.x = 0.0f; m.y = 0.0f; m.z = 0.0f; m.w = 0.0f;

#pragma unroll
    for (int t = 0; t < T_STEPS; ++t) {
        v4f spike;
#pragma unroll
        for (int c = 0; c < 4; ++c) {
            const float mm   = fmaf(m[c], TAU, xv[t][c]);  // mem = mem*tau + x
            const bool  fire = (mm >= THRESH);
            spike[c] = fire ? 1.0f : 0.0f;                 // spike output
            m[c]     = fire ? 0.0f : mm;                   // hard reset
        }
        __builtin_nontemporal_store(spike, &out[base + (size_t)t * SVEC]);
    }
}

extern "C" void kernel_launch(void* const* d_in, const int* in_sizes, int n_in,
                              void* d_out, int out_size, void* d_ws, size_t ws_size,
                              hipStream_t stream) {
    const v4f* x   = (const v4f*)d_in[0];
    v4f*       out = (v4f*)d_out;

    // in_sizes[0] = B*T*C*H*W = 33,554,432; each thread handles T_STEPS float4s.
    const int nthreads = in_sizes[0] / (4 * T_STEPS);      // 1,048,576
    const int block    = 256;                              // 8 wave32 per block
    const int grid     = (nthreads + block - 1) / block;   // 4096

    lif_scan_kernel<<<grid, block, 0, stream>>>(x, out, nthreads);
}